// CellAttention_47115791237126
// MI455X (gfx1250) — compile-verified
//
#include <hip/hip_runtime.h>
#include <hip/hip_bf16.h>

typedef __attribute__((ext_vector_type(2))) float v2f;
typedef __attribute__((ext_vector_type(8))) float v8f;

// ---------------------------------------------------------------------------
// Dual-GEMM projection: Out[m,a] = sum_k X1[m,k]*W1[a,k] (+ X2[m,k]*W2[a,k])
//                                  + b1[a] (+ b2[a])
// One wave computes one 16x16 output tile with V_WMMA_F32_16X16X4_F32.
// A layout (16x4 f32): lanes 0-15 -> M=lane, VGPR{0,1} = K{k,k+1};
//                      lanes 16-31 -> M=lane-16, VGPR{0,1} = K{k+2,k+3}.
// B layout (4x16 f32): lanes 0-15 -> N=lane, VGPR{0,1} = K{k,k+1};
//                      lanes 16-31 -> N=lane-16, VGPR{0,1} = K{k+2,k+3}.
// C layout: VGPR v, lanes 0-15 -> M=v; lanes 16-31 -> M=v+8.
// ---------------------------------------------------------------------------
__global__ __launch_bounds__(256) void proj_wmma_kernel(
    const float* __restrict__ X1, const float* __restrict__ W1,
    const float* __restrict__ X2, const float* __restrict__ W2,
    const float* __restrict__ b1, const float* __restrict__ b2,
    float* __restrict__ Out, int M, int Adim, int K)
{
    const int wave  = (blockIdx.x * blockDim.x + threadIdx.x) >> 5;
    const int lane  = threadIdx.x & 31;
    const int tilesA = Adim >> 4;
    const int m0 = (wave / tilesA) << 4;
    const int a0 = (wave % tilesA) << 4;
    if (m0 >= M) return;                     // wave-uniform (EXEC stays full)

    const int l15   = lane & 15;
    const int khalf = (lane >> 4) << 1;      // 0 or 2

    v8f c = {};
    {
        const float* xr = X1 + (size_t)(m0 + l15) * K + khalf;
        const float* wr = W1 + (size_t)(a0 + l15) * K + khalf;
        for (int k = 0; k < K; k += 4) {
            v2f a = *(const v2f*)(xr + k);
            v2f b = *(const v2f*)(wr + k);
            c = __builtin_amdgcn_wmma_f32_16x16x4_f32(
                    false, a, false, b, (short)0, c, false, false);
        }
    }
    if (X2) {
        const float* xr = X2 + (size_t)(m0 + l15) * K + khalf;
        const float* wr = W2 + (size_t)(a0 + l15) * K + khalf;
        for (int k = 0; k < K; k += 4) {
            v2f a = *(const v2f*)(xr + k);
            v2f b = *(const v2f*)(wr + k);
            c = __builtin_amdgcn_wmma_f32_16x16x4_f32(
                    false, a, false, b, (short)0, c, false, false);
        }
    }

    float bias = b1[a0 + l15];
    if (b2) bias += b2[a0 + l15];

    const int rbase = m0 + ((lane >> 4) << 3);
    float* outp = Out + (size_t)rbase * Adim + a0 + l15;
#pragma unroll
    for (int v = 0; v < 8; ++v)
        outp[(size_t)v * Adim] = c[v] + bias;
}

// ---------------------------------------------------------------------------
// Fused score + softmax. One block (8 wave32) per row n.
// score[n,p] = sum_a relu(att1[p,a] + s[n,a]) * Wf[a] + bf
// h (N*P*ATT = 256M elements) is never materialized.
// Each wave handles one p at a time: 32 lanes x 8 ATT values (ATT=256),
// coalesced float4 loads, then a 5-step wave32 xor-shuffle reduction.
// Row of P=1024 scores lives in LDS for the softmax.
// ---------------------------------------------------------------------------
__global__ __launch_bounds__(256) void score_softmax_kernel(
    const float* __restrict__ att1,   // (P, ATT)
    const float* __restrict__ sbuf,   // (N, ATT)
    const float* __restrict__ Wf,     // (ATT)
    const float* __restrict__ bf,     // (1)
    float* __restrict__ alpha,        // (N, P)
    int P, int ATT)
{
    const int n    = blockIdx.x;
    const int tid  = threadIdx.x;
    const int lane = tid & 31;
    const int wid  = tid >> 5;       // 0..7

    __shared__ float sc[1024];
    __shared__ float wred[8];

    // Per-lane resident slices of s[n,:] and Wf (8 contiguous values each).
    const float4* s4 = (const float4*)(sbuf + (size_t)n * ATT) + lane * 2;
    const float4* w4 = (const float4*)Wf + lane * 2;
    const float4 s0 = s4[0], s1 = s4[1];
    const float4 w0 = w4[0], w1 = w4[1];
    const float bfv = bf[0];

    for (int p = wid; p < P; p += 8) {
        const float4* a4 = (const float4*)(att1 + (size_t)p * ATT) + lane * 2;
        const float4 x0 = a4[0], x1 = a4[1];
        float acc;
        acc = fmaxf(x0.x + s0.x, 0.f) * w0.x;
        acc = fmaf(fmaxf(x0.y + s0.y, 0.f), w0.y, acc);
        acc = fmaf(fmaxf(x0.z + s0.z, 0.f), w0.z, acc);
        acc = fmaf(fmaxf(x0.w + s0.w, 0.f), w0.w, acc);
        acc = fmaf(fmaxf(x1.x + s1.x, 0.f), w1.x, acc);
        acc = fmaf(fmaxf(x1.y + s1.y, 0.f), w1.y, acc);
        acc = fmaf(fmaxf(x1.z + s1.z, 0.f), w1.z, acc);
        acc = fmaf(fmaxf(x1.w + s1.w, 0.f), w1.w, acc);
#pragma unroll
        for (int off = 16; off >= 1; off >>= 1)
            acc += __shfl_xor(acc, off, 32);
        if (lane == 0) sc[p] = acc + bfv;
    }
    __syncthreads();

    // ---- softmax over sc[0..P) ----
    float m = -3.402823466e38f;
    for (int p = tid; p < P; p += 256) m = fmaxf(m, sc[p]);
#pragma unroll
    for (int off = 16; off >= 1; off >>= 1)
        m = fmaxf(m, __shfl_xor(m, off, 32));
    if (lane == 0) wred[wid] = m;
    __syncthreads();
    if (wid == 0) {
        float mm = wred[lane & 7];
#pragma unroll
        for (int off = 4; off >= 1; off >>= 1)
            mm = fmaxf(mm, __shfl_xor(mm, off, 32));
        if (lane == 0) wred[0] = mm;
    }
    __syncthreads();
    const float rowmax = wred[0];
    __syncthreads();

    float ssum = 0.f;
    for (int p = tid; p < P; p += 256) {
        float e = __expf(sc[p] - rowmax);
        sc[p] = e;
        ssum += e;
    }
#pragma unroll
    for (int off = 16; off >= 1; off >>= 1)
        ssum += __shfl_xor(ssum, off, 32);
    if (lane == 0) wred[wid] = ssum;
    __syncthreads();
    if (wid == 0) {
        float t = wred[lane & 7];
#pragma unroll
        for (int off = 4; off >= 1; off >>= 1)
            t += __shfl_xor(t, off, 32);
        if (lane == 0) wred[0] = t;
    }
    __syncthreads();
    const float inv = 1.0f / wred[0];

    float* arow = alpha + (size_t)n * P;
    for (int p = tid; p < P; p += 256) arow[p] = sc[p] * inv;
}

// ---------------------------------------------------------------------------
// awe = alpha (N x P) @ enc (P x E), f32 WMMA 16x16x4. One wave per 16x16 tile.
// ---------------------------------------------------------------------------
__global__ __launch_bounds__(256) void awe_wmma_kernel(
    const float* __restrict__ alpha, const float* __restrict__ enc,
    float* __restrict__ awe, int N, int E, int P)
{
    const int wave = (blockIdx.x * blockDim.x + threadIdx.x) >> 5;
    const int lane = threadIdx.x & 31;
    const int tilesE = E >> 4;
    const int n0 = (wave / tilesE) << 4;
    const int e0 = (wave % tilesE) << 4;
    if (n0 >= N) return;                    // wave-uniform

    const int l15   = lane & 15;
    const int khalf = (lane >> 4) << 1;

    v8f c = {};
    const float* arow = alpha + (size_t)(n0 + l15) * P + khalf;
    const float* bcol = enc + e0 + l15;     // B[k][e] = enc[k*E + e]
    for (int k = 0; k < P; k += 4) {
        v2f a = *(const v2f*)(arow + k);
        v2f b;
        b.x = bcol[(size_t)(k + khalf) * E];
        b.y = bcol[(size_t)(k + khalf + 1) * E];
        c = __builtin_amdgcn_wmma_f32_16x16x4_f32(
                false, a, false, b, (short)0, c, false, false);
    }

    const int rbase = n0 + ((lane >> 4) << 3);
    float* outp = awe + (size_t)rbase * E + e0 + l15;
#pragma unroll
    for (int v = 0; v < 8; ++v)
        outp[(size_t)v * E] = c[v];
}

// ---------------------------------------------------------------------------
extern "C" void kernel_launch(void* const* d_in, const int* in_sizes, int n_in,
                              void* d_out, int out_size, void* d_ws, size_t ws_size,
                              hipStream_t stream) {
    constexpr int P = 1024, N = 1024, ENC = 512, ATT = 256, K = 512;

    const float* enc  = (const float*)d_in[0];   // (1,P,ENC)
    const float* dec  = (const float*)d_in[1];   // (N,TAG)
    const float* lang = (const float*)d_in[2];   // (N,LANG)
    const float* We   = (const float*)d_in[3];   // (ATT,ENC)
    const float* be   = (const float*)d_in[4];
    const float* Wt   = (const float*)d_in[5];
    const float* bt   = (const float*)d_in[6];
    const float* Wl   = (const float*)d_in[7];
    const float* bl   = (const float*)d_in[8];
    const float* Wf   = (const float*)d_in[9];   // (1,ATT)
    const float* bf   = (const float*)d_in[10];  // (1,)

    float* att1 = (float*)d_ws;                          // P*ATT  = 1 MB
    float* sbuf = att1 + (size_t)P * ATT;                // N*ATT  = 1 MB

    float* awe_out   = (float*)d_out;                    // N*ENC
    float* alpha_out = awe_out + (size_t)N * ENC;        // N*P

    // att1 = enc @ We^T + be          : (P/16)*(ATT/16) = 1024 waves
    {
        int waves = (P / 16) * (ATT / 16);
        proj_wmma_kernel<<<waves * 32 / 256, 256, 0, stream>>>(
            enc, We, nullptr, nullptr, be, nullptr, att1, P, ATT, K);
    }
    // s = dec @ Wt^T + lang @ Wl^T + bt + bl
    {
        int waves = (N / 16) * (ATT / 16);
        proj_wmma_kernel<<<waves * 32 / 256, 256, 0, stream>>>(
            dec, Wt, lang, Wl, bt, bl, sbuf, N, ATT, K);
    }
    // fused scores + softmax -> alpha
    score_softmax_kernel<<<N, 256, 0, stream>>>(att1, sbuf, Wf, bf,
                                                alpha_out, P, ATT);
    // awe = alpha @ enc               : (N/16)*(ENC/16) = 2048 waves
    {
        int waves = (N / 16) * (ENC / 16);
        awe_wmma_kernel<<<waves * 32 / 256, 256, 0, stream>>>(
            alpha_out, enc, awe_out, N, ENC, P);
    }
}